// QuadPool_16458314678351
// MI455X (gfx1250) — compile-verified
//
#include <hip/hip_runtime.h>

// QuadPool segment-max for MI455X (gfx1250).
// Memory-bound: reads features once (1.02 GB, NT async HBM->LDS stream),
// writes pooled once (0.31 GB, NT b128 stores) => ~58 us roofline @ 23.3 TB/s.
// One wave32 per parent; children are a contiguous run in the sorted key
// array. Data path: global_load_async_to_lds_b128 (ASYNCcnt) -> ds_load_b128
// -> v_max_num_f32 -> non-temporal global_store_b128.

typedef float v4f __attribute__((ext_vector_type(4)));

#define QP_C           256          // channels (reference fixes C=256)
#define QP_ROW_BYTES   1024         // 256 * 4B
#define QP_WAVES       4            // waves per block
#define QP_NBUF        8            // 1KB row slots per wave (8KB/wave LDS)

static __device__ __forceinline__ v4f vmax4(v4f a, v4f b) {
    a.x = fmaxf(a.x, b.x); a.y = fmaxf(a.y, b.y);
    a.z = fmaxf(a.z, b.z); a.w = fmaxf(a.w, b.w);
    return a;
}

__global__ void __launch_bounds__(QP_WAVES * 32)
quadpool_segmax_kernel(const float* __restrict__ features,
                       const int*   __restrict__ keys,
                       float*       __restrict__ pooled,
                       int N, int P)
{
    __shared__ __align__(16) unsigned char buf[QP_WAVES][QP_NBUF][QP_ROW_BYTES];

    const int lane = threadIdx.x & 31;
    const int wave = threadIdx.x >> 5;
    int p = blockIdx.x * QP_WAVES + wave;            // wave-uniform
    if (p >= P) return;                              // whole wave exits together
    p = __builtin_amdgcn_readfirstlane(p);           // make uniformity provable
                                                     // -> scalar search path

    const int v0 = p << 2;                           // children: key in [4p, 4p+4)
    const int v1 = v0 + 4;

    // ---- scalar binary search for run start (keys: 4MB, L2-resident)
    int lo = 0, hi = N;
    while (lo < hi) {
        int mid = (lo + hi) >> 1;
        if (keys[mid] < v0) lo = mid + 1; else hi = mid;
    }
    const int start = lo;

    // ---- vectorized run-length scan: lane l probes keys[start+l].
    // keys sorted => ballot mask is a prefix; one coalesced load suffices
    // unless a parent has >= 32 children (Poisson(3.3): ~never, loop is exact).
    int end = start;
    for (;;) {
        const int i = end + lane;
        const int k = (i < N) ? keys[i] : 0x7fffffff;
        const unsigned m = __builtin_amdgcn_ballot_w32(k < v1);
        const int c = __popc(m);
        end += c;
        if (c < 32) break;
    }

    // ---- accumulators: lane owns floats [4l,4l+4) and [128+4l, +4)
    v4f a0 = { -__builtin_inff(), -__builtin_inff(),
               -__builtin_inff(), -__builtin_inff() };
    v4f a1 = a0;

    const unsigned lds_lane =
        (unsigned)(size_t)(&buf[wave][0][0]) + (unsigned)(lane * 16);
    const unsigned long long gbase =
        (unsigned long long)(size_t)features + (unsigned)(lane * 16);

    for (int base = start; base < end; base += QP_NBUF) {
        const int cnt = (end - base < QP_NBUF) ? (end - base) : QP_NBUF;

        // async HBM->LDS: 2 x b128 per 1KB row (32 lanes x 16B = 512B each);
        // NT hint: features are read exactly once, keep L2 for keys/pooled.
        for (int j = 0; j < cnt; ++j) {
            const unsigned long long g =
                gbase + (unsigned long long)(base + j) * (unsigned long long)QP_ROW_BYTES;
            const unsigned l = lds_lane + (unsigned)(j * QP_ROW_BYTES);
            asm volatile("global_load_async_to_lds_b128 %0, %1, off th:TH_LOAD_NT"
                         :: "v"(l), "v"(g) : "memory");
            // INST_OFFSET is added to both LDS and memory addresses
            asm volatile("global_load_async_to_lds_b128 %0, %1, off offset:512 th:TH_LOAD_NT"
                         :: "v"(l), "v"(g) : "memory");
        }

        asm volatile("s_wait_asynccnt 0x0" ::: "memory");

        // reduce staged rows out of LDS (ds_load_b128 pairs, no bank conflicts)
        for (int j = 0; j < cnt; ++j) {
            const unsigned char* s = &buf[wave][j][0] + lane * 16;
            a0 = vmax4(a0, *(const v4f*)(s));
            a1 = vmax4(a1, *(const v4f*)(s + 512));
        }
    }

    if (end <= start) {                 // empty segment: reference fills with 0
        a0 = (v4f){0.f, 0.f, 0.f, 0.f};
        a1 = a0;
    }

    // pooled written once, never re-read: non-temporal b128 stores
    float* dst = pooled + (size_t)p * QP_C + lane * 4;
    __builtin_nontemporal_store(a0, (v4f*)(dst));        // channels [4l, 4l+4)
    __builtin_nontemporal_store(a1, (v4f*)(dst + 128));  // channels [128+4l, +4)
}

__global__ void __launch_bounds__(256)
quadpool_pidx_kernel(const int* __restrict__ keys, float* __restrict__ out, int N)
{
    const int i = blockIdx.x * blockDim.x + threadIdx.x;
    if (i < N)
        __builtin_nontemporal_store((float)(keys[i] >> 2), out + i);
}

extern "C" void kernel_launch(void* const* d_in, const int* in_sizes, int n_in,
                              void* d_out, int out_size, void* d_ws, size_t ws_size,
                              hipStream_t stream)
{
    (void)n_in; (void)out_size; (void)d_ws; (void)ws_size;

    const float* features = (const float*)d_in[0];   // [N, 256] f32
    const int*   keys     = (const int*)d_in[1];     // [N] sorted, < 4P
    const int    N        = in_sizes[1];
    const int    P        = in_sizes[2];             // parent_level_keys = arange(P)

    float* pooled = (float*)d_out;                       // [P, 256]
    float* pidx   = (float*)d_out + (size_t)P * QP_C;    // [N] parent indices (as f32)

    const int blocks = (P + QP_WAVES - 1) / QP_WAVES;
    quadpool_segmax_kernel<<<blocks, QP_WAVES * 32, 0, stream>>>(features, keys, pooled, N, P);
    quadpool_pidx_kernel<<<(N + 255) / 256, 256, 0, stream>>>(keys, pidx, N);
}